// Compensation_20220706029991
// MI455X (gfx1250) — compile-verified
//
#include <hip/hip_runtime.h>
#include <hip/hip_bf16.h>

#define DFEAT 256
#define LSTRIDE 260            // 256 + 4 pad -> conflict-free b64 LDS reads (260 % 64 == 4)
#define ASIZE   (16 * LSTRIDE) // one A strip in LDS (floats)

typedef __attribute__((ext_vector_type(2))) float v2f;
typedef __attribute__((ext_vector_type(8))) float v8f;

// ---------------- degree / normalization ----------------

__global__ __launch_bounds__(256) void k_deg_init(float* __restrict__ deg, int n) {
    int i = blockIdx.x * 256 + threadIdx.x;
    if (i < n) deg[i] = 1.0f;   // self-loop contributes 1
}

__global__ __launch_bounds__(256) void k_deg_accum(const int* __restrict__ dst,
                                                   float* __restrict__ deg, int e) {
    int i = blockIdx.x * 256 + threadIdx.x;
    if (i < e) atomicAdd(&deg[dst[i]], 1.0f);
}

__global__ __launch_bounds__(256) void k_dinv(const float* __restrict__ deg,
                                              float* __restrict__ dinv, int n) {
    int i = blockIdx.x * 256 + threadIdx.x;
    if (i < n) dinv[i] = rsqrtf(deg[i]);
}

// ---- CDNA5 async global->LDS DMA (ASYNCcnt), no VGPR staging ----
// GVS mode: saddr = SGPR base, vaddr = 32-bit byte offset, vdst = 32-bit LDS byte address.
__device__ __forceinline__ void async_ld_b128(unsigned lds_addr, unsigned gbyte_off,
                                              const float* gbase) {
    asm volatile("global_load_async_to_lds_b128 %0, %1, %2"
                 :: "v"(lds_addr), "v"(gbyte_off), "s"(gbase)
                 : "memory");
}
__device__ __forceinline__ void wait_async0() {
    asm volatile("s_wait_asynccnt 0x0" ::: "memory");
}

// ---------------- persistent WMMA f32 GEMM ----------------
// W staged (transposed) in LDS once per block; A strips double-buffered in LDS via
// async DMA overlapped with the WMMA loop. 8 waves/block, 2 n-tiles per wave.
// mode 1: outT = dropout(relu(acc + bias))                        (layer 0)
// mode 2: outT = acc (raw), out2 = bias + acc*dinv[m]^2           (conv pre-scatter, self-loop)
// LDS: Wt[256][LSTRIDE] + 2 x Asm[16][LSTRIDE] = 299,520 B < 320 KB/WGP

__global__ __launch_bounds__(256) void k_gemm_wmma(const float* __restrict__ A,
                                                   const float* __restrict__ W,
                                                   const float* __restrict__ bias,
                                                   const unsigned char* __restrict__ mask,
                                                   const float* __restrict__ dinv,
                                                   float* __restrict__ outT,
                                                   float* __restrict__ out2,
                                                   int n_rows, int mode) {
    extern __shared__ float smem[];
    float* Wt  = smem;                       // [256][LSTRIDE], Wt[n][k] = W[k][n]
    float* Asm = smem + DFEAT * LSTRIDE;     // 2 x [16][LSTRIDE]

    const int tid  = threadIdx.x;
    const int lane = tid & 31;
    const int wave = tid >> 5;               // 0..7
    const int ml   = lane & 15;              // A-row / B,C,D-column within tile
    const int kh   = lane >> 4;              // K-pair select / M-half of C

    // per-thread A-staging coordinates: 4 x 16B chunks, rows sr0+4i, col sc
    const int sr0 = tid >> 6;
    const int sc  = (tid & 63) << 2;
    // 32-bit LDS byte address of the A buffers (flat addr low 32 bits == DS offset)
    const unsigned lds_a = (unsigned)(uintptr_t)Asm;

    // ---- kick off async DMA of the first A strip into buffer 0 ----
    const int tiles_m = n_rows / 16;
    const int mt0 = blockIdx.x;
    if (mt0 < tiles_m) {
#pragma unroll
        for (int i = 0; i < 4; ++i) {
            const int r = sr0 + 4 * i;
            async_ld_b128(lds_a + (unsigned)((r * LSTRIDE + sc) * 4),
                          (unsigned)(((mt0 * 16 + r) * DFEAT + sc) * 4), A);
        }
    }

    // ---- stage W transposed into LDS (once per block, overlaps the DMA) ----
#pragma unroll 4
    for (int i = 0; i < 64; ++i) {
        int f  = tid + 256 * i;              // 0..16383
        int k  = f >> 6;                     // 0..255
        int n4 = (f & 63) << 2;              // 0..252
        float4 w = *(const float4*)(W + (size_t)k * DFEAT + n4);
        Wt[(n4 + 0) * LSTRIDE + k] = w.x;
        Wt[(n4 + 1) * LSTRIDE + k] = w.y;
        Wt[(n4 + 2) * LSTRIDE + k] = w.z;
        Wt[(n4 + 3) * LSTRIDE + k] = w.w;
    }

    const int n0a = 32 * wave;               // this wave's two column tiles
    const int n0b = n0a + 16;
    const float* B0 = Wt + (n0a + ml) * LSTRIDE;
    const float* B1 = Wt + (n0b + ml) * LSTRIDE;

    int buf = 0;
    for (int mt = mt0; mt < tiles_m; mt += gridDim.x) {
        const int m0 = mt * 16;
        wait_async0();                       // my DMA into Asm[buf] landed
        __syncthreads();                     // everyone's landed; Asm[buf^1] free

        // ---- issue async DMA for the next strip into the other buffer ----
        const int mtn = mt + gridDim.x;
        if (mtn < tiles_m) {
            const unsigned dstb = lds_a + (unsigned)(((buf ^ 1) * ASIZE) * 4);
#pragma unroll
            for (int i = 0; i < 4; ++i) {
                const int r = sr0 + 4 * i;
                async_ld_b128(dstb + (unsigned)((r * LSTRIDE + sc) * 4),
                              (unsigned)(((mtn * 16 + r) * DFEAT + sc) * 4), A);
            }
        }

        // ---- 64 x 2 WMMAs, all operands from LDS (DMA streams underneath) ----
        const float* Arow = Asm + buf * ASIZE + ml * LSTRIDE;
        v8f acc0 = {};
        v8f acc1 = {};
#pragma unroll 8
        for (int k = 0; k < DFEAT; k += 4) {
            const int ka = k + 2 * kh;
            v2f a  = *(const v2f*)(Arow + ka);   // A 16x4 frag (ISA f32 layout)
            v2f b0 = *(const v2f*)(B0 + ka);     // B 4x16 frag, col n0a+ml
            v2f b1 = *(const v2f*)(B1 + ka);     // B 4x16 frag, col n0b+ml
            acc0 = __builtin_amdgcn_wmma_f32_16x16x4_f32(false, a, false, b0,
                                                         (short)0, acc0, false, false);
            acc1 = __builtin_amdgcn_wmma_f32_16x16x4_f32(false, a, false, b1,
                                                         (short)0, acc1, false, false);
        }
        buf ^= 1;

        // ---- epilogue ----
        const float bva = bias[n0a + ml];
        const float bvb = bias[n0b + ml];
#pragma unroll
        for (int r = 0; r < 8; ++r) {
            const int m = m0 + r + 8 * kh;   // C/D layout: VGPR r -> row r (+8 hi half)
            float va = acc0[r];
            float vb = acc1[r];
            if (mode == 1) {                 // bias + relu + dropout(p=0.5)
                va += bva;                   vb += bvb;
                va = va > 0.0f ? va : 0.0f;  vb = vb > 0.0f ? vb : 0.0f;
                va = mask[(size_t)m * DFEAT + n0a + ml] ? va * 2.0f : 0.0f;
                vb = mask[(size_t)m * DFEAT + n0b + ml] ? vb * 2.0f : 0.0f;
                outT[(size_t)m * DFEAT + n0a + ml] = va;
                outT[(size_t)m * DFEAT + n0b + ml] = vb;
            } else {                         // mode 2: raw T + self-loop-initialized acc
                outT[(size_t)m * DFEAT + n0a + ml] = va;
                outT[(size_t)m * DFEAT + n0b + ml] = vb;
                const float dv = dinv[m];
                const float id = dv * dv;    // self-loop norm = dinv^2 (matches reference)
                out2[(size_t)m * DFEAT + n0a + ml] = bva + va * id;
                out2[(size_t)m * DFEAT + n0b + ml] = bvb + vb * id;
            }
        }
    }
}

// ---------------- edge scatter: out2 += norm * T[src] ----------------

__global__ __launch_bounds__(256) void k_scatter(const int* __restrict__ src,
                                                 const int* __restrict__ dst,
                                                 const float* __restrict__ dinv,
                                                 const float* __restrict__ T,
                                                 float* __restrict__ out, int e) {
    long long t = (long long)blockIdx.x * 256 + threadIdx.x;
    int edge = (int)(t >> 6);                // 64 threads per edge (wave-uniform)
    int q    = (int)(t & 63);                // 4 floats per thread
    if (edge >= e) return;
    int s = src[edge];
    int d = dst[edge];
    float w = dinv[s] * dinv[d];
    float4 h = *(const float4*)(T + (size_t)s * DFEAT + q * 4);
    float* o = out + (size_t)d * DFEAT + q * 4;
    atomicAdd(o + 0, h.x * w);
    atomicAdd(o + 1, h.y * w);
    atomicAdd(o + 2, h.z * w);
    atomicAdd(o + 3, h.w * w);
}

// ---------------- elementwise ----------------

__global__ __launch_bounds__(256) void k_relu_drop(float* __restrict__ x,
                                                   const unsigned char* __restrict__ mask,
                                                   int total) {
    int t = blockIdx.x * 256 + threadIdx.x;
    if (t < total) {
        float v = x[t];
        v = v > 0.0f ? v : 0.0f;
        x[t] = mask[t] ? v * 2.0f : 0.0f;
    }
}

__global__ __launch_bounds__(256) void k_final(const float* __restrict__ z,
                                               const float* __restrict__ h,
                                               float* __restrict__ out, int total) {
    int t = blockIdx.x * 256 + threadIdx.x;
    if (t < total) out[t] = z[t] * h[t];
}

// ---------------- host orchestration ----------------

extern "C" void kernel_launch(void* const* d_in, const int* in_sizes, int n_in,
                              void* d_out, int out_size, void* d_ws, size_t ws_size,
                              hipStream_t stream) {
    const float* z      = (const float*)d_in[0];
    const float* omega  = (const float*)d_in[1];
    const int*   eidx   = (const int*)  d_in[2];   // [2, E]
    const float* W_lin  = (const float*)d_in[3];
    const float* b_lin  = (const float*)d_in[4];
    const float* W_g1   = (const float*)d_in[5];
    const float* b_g1   = (const float*)d_in[6];
    const float* W_g2   = (const float*)d_in[7];
    const float* b_g2   = (const float*)d_in[8];
    const unsigned char* mask1 = (const unsigned char*)d_in[9];
    const unsigned char* mask2 = (const unsigned char*)d_in[10];
    float* out = (float*)d_out;

    const int N  = in_sizes[0] / DFEAT;            // 100000
    const int E  = in_sizes[2] / 2;                // 1600000
    const int NT = N * DFEAT;                      // 25.6M

    const int* src = eidx;
    const int* dst = eidx + E;

    // workspace layout (f32)
    float* buf0 = (float*)d_ws;                    // N x D
    float* buf1 = buf0 + (size_t)NT;               // N x D
    float* buf2 = buf1 + (size_t)NT;               // N x D
    float* deg  = buf2 + (size_t)NT;               // N
    float* dinv = deg + N;                         // N

    const int eb  = (E + 255) / 256;
    const int nb  = (N + 255) / 256;
    const int ntb = (NT + 255) / 256;
    const int tiles_m = N / 16;                    // 6250
    const int gemm_blocks = tiles_m < 625 ? tiles_m : 625;   // persistent, 10 tiles each
    const size_t shmem = (size_t)(DFEAT * LSTRIDE + 2 * ASIZE) * sizeof(float); // 299,520 B
    const int scat_blocks = (int)(((long long)E * 64 + 255) / 256);

    // normalization
    k_deg_init <<<nb, 256, 0, stream>>>(deg, N);
    k_deg_accum<<<eb, 256, 0, stream>>>(dst, deg, E);
    k_dinv     <<<nb, 256, 0, stream>>>(deg, dinv, N);

    // layer 0: h0 = dropout(relu(omega @ W_lin + b_lin))
    k_gemm_wmma<<<gemm_blocks, 256, shmem, stream>>>(omega, W_lin, b_lin, mask1,
                                                     nullptr, buf0, nullptr, N, 1);

    // layer 1: T = h0 @ W_g1 ; acc = b + T*dinv^2 (self-loop) ; scatter ; relu+drop
    k_gemm_wmma<<<gemm_blocks, 256, shmem, stream>>>(buf0, W_g1, b_g1, nullptr,
                                                     dinv, buf1, buf2, N, 2);
    k_scatter  <<<scat_blocks, 256, 0, stream>>>(src, dst, dinv, buf1, buf2, E);
    k_relu_drop<<<ntb, 256, 0, stream>>>(buf2, mask2, NT);

    // layer 2: T = h1 @ W_g2 ; acc = b + T*dinv^2 ; scatter
    k_gemm_wmma<<<gemm_blocks, 256, shmem, stream>>>(buf2, W_g2, b_g2, nullptr,
                                                     dinv, buf1, buf0, N, 2);
    k_scatter  <<<scat_blocks, 256, 0, stream>>>(src, dst, dinv, buf1, buf0, E);

    // out = z * h
    k_final<<<ntb, 256, 0, stream>>>(z, buf0, out, NT);
}